// NetSat1_85504208928810
// MI455X (gfx1250) — compile-verified
//
#include <hip/hip_runtime.h>
#include <stdint.h>

#define C_DIM   1000
#define BLOCK   256
#define NWAVES  (BLOCK / 32)
#define CHUNKS  250            // float4 chunks per row (1000 floats)

typedef __attribute__((ext_vector_type(4))) float v4f;

__device__ __forceinline__ void top2_merge(float& m1, float& m2, int& i1,
                                           float bm1, float bm2, int bi1) {
    if (bm1 > m1) {
        m2 = fmaxf(bm2, m1);
        m1 = bm1;
        i1 = bi1;
    } else {
        m2 = fmaxf(m2, bm1);
    }
}

// scan 4 consecutive elements (base index c0) into a running top-2 + argmax
__device__ __forceinline__ void top2_scan4(const v4f& x, int c0,
                                           float& m1, float& m2, int& i1) {
    #pragma unroll
    for (int k = 0; k < 4; ++k) {
        float v = x[k];
        if (v > m1)      { m2 = m1; m1 = v; i1 = c0 + k; }
        else if (v > m2) { m2 = v; }
    }
}

__global__ __launch_bounds__(BLOCK) void loo_minmax_kernel(
        const float* __restrict__ y1,
        const float* __restrict__ y2,
        float* __restrict__ out) {
    __shared__ __align__(16) float srow[2 * C_DIM];  // [0,1000)=y1 row, [1000,2000)=y2 row
    __shared__ float wm1[2][NWAVES], wm2[2][NWAVES];
    __shared__ int   wix[2][NWAVES];
    __shared__ float bt1[2], bt2[2];
    __shared__ int   bix[2];
    __shared__ float wmax[NWAVES];

    const int tid    = threadIdx.x;
    const int lane   = tid & 31;
    const int wave   = tid >> 5;
    const int row    = blockIdx.x;
    const bool active = (tid < CHUNKS);

    // ---- Stage both rows into LDS via CDNA5 async global->LDS B128 copies ----
    // Row stride is 4000 B (250*16), so every row base stays 16-byte aligned.
    const uint32_t lds_base = (uint32_t)(uintptr_t)&srow[0];
    if (active) {
        unsigned long long ga1 =
            (unsigned long long)(uintptr_t)(y1 + (size_t)row * C_DIM) + (uint32_t)tid * 16u;
        unsigned long long ga2 =
            (unsigned long long)(uintptr_t)(y2 + (size_t)row * C_DIM) + (uint32_t)tid * 16u;
        uint32_t d1 = lds_base + (uint32_t)tid * 16u;
        uint32_t d2 = d1 + (uint32_t)(CHUNKS * 16);
        asm volatile("global_load_async_to_lds_b128 %0, %1, off"
                     :: "v"(d1), "v"(ga1) : "memory");
        asm volatile("global_load_async_to_lds_b128 %0, %1, off"
                     :: "v"(d2), "v"(ga2) : "memory");
    }
    asm volatile("s_wait_asynccnt 0" ::: "memory");
    __syncthreads();

    // ---- One LDS read per row per thread; values stay in registers ----
    const float NEG_INF = -__builtin_inff();
    const int c0 = tid * 4;
    v4f xa = {NEG_INF, NEG_INF, NEG_INF, NEG_INF};
    v4f xb = {NEG_INF, NEG_INF, NEG_INF, NEG_INF};
    if (active) {
        xa = ((const v4f*)srow)[tid];           // ds_load_b128
        xb = ((const v4f*)srow)[CHUNKS + tid];  // ds_load_b128
    }

    // ---- Pass 1: per-thread top-2 (+ argmax) for both rows ----
    float am1 = NEG_INF, am2 = NEG_INF; int ai = -1;
    float bm1 = NEG_INF, bm2 = NEG_INF; int bi = -1;
    top2_scan4(xa, c0, am1, am2, ai);
    top2_scan4(xb, c0, bm1, bm2, bi);

    // wave32 xor-shuffle merge
    #pragma unroll
    for (int off = 16; off > 0; off >>= 1) {
        float oa1 = __shfl_xor(am1, off, 32);
        float oa2 = __shfl_xor(am2, off, 32);
        int   oai = __shfl_xor(ai,  off, 32);
        top2_merge(am1, am2, ai, oa1, oa2, oai);
        float ob1 = __shfl_xor(bm1, off, 32);
        float ob2 = __shfl_xor(bm2, off, 32);
        int   obi = __shfl_xor(bi,  off, 32);
        top2_merge(bm1, bm2, bi, ob1, ob2, obi);
    }
    if (lane == 0) {
        wm1[0][wave] = am1; wm2[0][wave] = am2; wix[0][wave] = ai;
        wm1[1][wave] = bm1; wm2[1][wave] = bm2; wix[1][wave] = bi;
    }
    __syncthreads();

    if (tid == 0) {
        #pragma unroll
        for (int k = 0; k < 2; ++k) {
            float m1 = wm1[k][0], m2 = wm2[k][0]; int ix = wix[k][0];
            #pragma unroll
            for (int w = 1; w < NWAVES; ++w)
                top2_merge(m1, m2, ix, wm1[k][w], wm2[k][w], wix[k][w]);
            bt1[k] = m1; bt2[k] = m2; bix[k] = ix;
        }
    }
    __syncthreads();

    // ---- Pass 2: max_c min(y1 - loo1, y2 - loo2), from registers ----
    const float t1a = bt1[0], t2a = bt2[0]; const int ia = bix[0];
    const float t1b = bt1[1], t2b = bt2[1]; const int ib = bix[1];
    float mx = NEG_INF;
    if (active) {
        #pragma unroll
        for (int k = 0; k < 4; ++k) {
            int c = c0 + k;
            float v1 = xa[k] - ((c == ia) ? t2a : t1a);
            float v2 = xb[k] - ((c == ib) ? t2b : t1b);
            mx = fmaxf(mx, fminf(v1, v2));
        }
    }
    #pragma unroll
    for (int off = 16; off > 0; off >>= 1)
        mx = fmaxf(mx, __shfl_xor(mx, off, 32));
    if (lane == 0) wmax[wave] = mx;
    __syncthreads();

    if (tid == 0) {
        float r = wmax[0];
        #pragma unroll
        for (int w = 1; w < NWAVES; ++w) r = fmaxf(r, wmax[w]);
        out[row] = r;
    }
}

extern "C" void kernel_launch(void* const* d_in, const int* in_sizes, int n_in,
                              void* d_out, int out_size, void* d_ws, size_t ws_size,
                              hipStream_t stream) {
    const float* y1 = (const float*)d_in[0];
    const float* y2 = (const float*)d_in[1];
    float* out = (float*)d_out;
    const int B = in_sizes[0] / C_DIM;   // 16384 rows of 1000
    loo_minmax_kernel<<<B, BLOCK, 0, stream>>>(y1, y2, out);
}